// DeformConv_18811956757233
// MI455X (gfx1250) — compile-verified
//
#include <hip/hip_runtime.h>
#include <hip/hip_bf16.h>

typedef float v2f __attribute__((ext_vector_type(2)));
typedef float v8f __attribute__((ext_vector_type(8)));
typedef unsigned int v4u __attribute__((ext_vector_type(4)));
typedef int v8i __attribute__((ext_vector_type(8)));
typedef int v4i __attribute__((ext_vector_type(4)));

#define B_IMG 8
#define C_IN  256
#define H_DIM 64
#define W_DIM 64
#define HW    4096
#define O_OUT 256
#define NOFF  27
#define CK    2304   // C_IN * 9
#define NPIX  32     // pixel tile per block (kernel 2)
#define KC    36     // K-chunk: 4 channels * 9 taps
#define KSTR  36     // LDS row stride for A ([row][K]) and V ([pix][K]);
                     // K offsets are multiples of 4 -> 36*row mod 64 covers all
                     // 16 {4m} residues per half-wave, +ksel splits odd/even
                     // pairs: all 64 banks distinct -> conflict-free b64 frags.
                     // KSTR==KC also makes the A chunk a dense [256][36] tile,
                     // i.e. exactly one TDM descriptor.

// ---------------------------------------------------------------------------
// TDM: one-instruction 2D tile copy w_conv[256 x 36 floats, row stride 2304]
// -> contiguous LDS. Descriptor per CDNA5 ISA.
// 6-arg builtin (clang-23 / amdgpu-toolchain signature).
// ---------------------------------------------------------------------------
__device__ __forceinline__ void tdm_load_A_tile(const float* gsrc, unsigned lds_addr) {
    unsigned long long ga = (unsigned long long)(uintptr_t)gsrc;
    v4u g0;
    g0.x = 1u;                                   // count=1 (valid), user mode
    g0.y = lds_addr;                             // bits 63:32  : lds_addr
    g0.z = (unsigned)ga;                         // bits 95:64  : global_addr lo
    g0.w = (unsigned)(ga >> 32) | 0x80000000u;   // global_addr hi | type=2
    v8i g1;
    g1[0] = 0x00020000;                          // data_size=2 -> 4 bytes
    g1[1] = 0;                                   // no atomic barrier
    g1[2] = 0x00001000;                          // tensor_dim0 = 0x10000000 (no OOB)
    g1[3] = (KC << 16) | 0x00001000;             // tile_dim0=36 | tensor_dim1 hi
    g1[4] = O_OUT;                               // tile_dim1=256, tile_dim2=0
    g1[5] = CK;                                  // tensor_dim0_stride = 2304
    g1[6] = 0;
    g1[7] = 0;
    v4i gz4 = {0, 0, 0, 0};                      // 2D: groups 2/3 unused
    v8i gz8 = {0, 0, 0, 0, 0, 0, 0, 0};          // group 4 unused
    __builtin_amdgcn_tensor_load_to_lds(g0, g1, gz4, gz4, gz8, 0);
}

// ---------------------------------------------------------------------------
// Kernel 1: 3x3 offset conv, 256 -> 27 (padded 32) channels, fp32 WMMA.
// ---------------------------------------------------------------------------
__global__ __launch_bounds__(256) void dcn_offset_conv_wmma(
    const float* __restrict__ x, const float* __restrict__ w_off,
    const float* __restrict__ b_off, float* __restrict__ om)
{
    __shared__ float A_lds[32 * KSTR];        //  4608 B : w_off chunk (pad M)
    __shared__ float V_lds[256 * KSTR];       // 36864 B : im2col [pix][K]

    const int b       = blockIdx.x >> 4;      // 16 blocks / image
    const int pixBase = (blockIdx.x & 15) * 256;
    const int tid    = threadIdx.x;
    const int lane   = tid & 31;
    const int waveId = tid >> 5;

    const int ksel = (lane < 16) ? 0 : 2;
    const int nlan = lane & 15;

    const float* xb = x + (size_t)b * (C_IN * HW);

    v8f acc00 = {}, acc01 = {}, acc10 = {}, acc11 = {};

    for (int cBase = 0; cBase < C_IN; cBase += 4) {
        for (int idx = tid; idx < 32 * KC; idx += 256) {
            int o  = idx / KC;
            int kk = idx - o * KC;
            float v = (o < NOFF) ? w_off[(size_t)o * CK + cBase * 9 + kk] : 0.0f;
            A_lds[o * KSTR + kk] = v;
        }
        for (int idx = tid; idx < 256 * KC; idx += 256) {
            int kk = idx >> 8;
            int p  = idx & 255;
            int c  = cBase + kk / 9;
            int k  = kk - (kk / 9) * 9;
            int pp = pixBase + p;
            int py = pp >> 6, px = pp & 63;
            int ty = py + k / 3 - 1;
            int tx = px + k % 3 - 1;
            bool ok = (ty >= 0) & (ty < H_DIM) & (tx >= 0) & (tx < W_DIM);
            V_lds[p * KSTR + kk] = ok ? xb[c * HW + ty * W_DIM + tx] : 0.0f;
        }
        __syncthreads();

        const int pw = waveId * 32;
#pragma unroll
        for (int kk = 0; kk < KC; kk += 4) {
            v2f a0 = *(const v2f*)&A_lds[(nlan)      * KSTR + kk + ksel];
            v2f a1 = *(const v2f*)&A_lds[(16 + nlan) * KSTR + kk + ksel];
            v2f b0 = *(const v2f*)&V_lds[(pw + nlan)      * KSTR + kk + ksel];
            v2f b1 = *(const v2f*)&V_lds[(pw + 16 + nlan) * KSTR + kk + ksel];
            acc00 = __builtin_amdgcn_wmma_f32_16x16x4_f32(false, a0, false, b0, (short)0, acc00, false, false);
            acc01 = __builtin_amdgcn_wmma_f32_16x16x4_f32(false, a0, false, b1, (short)0, acc01, false, false);
            acc10 = __builtin_amdgcn_wmma_f32_16x16x4_f32(false, a1, false, b0, (short)0, acc10, false, false);
            acc11 = __builtin_amdgcn_wmma_f32_16x16x4_f32(false, a1, false, b1, (short)0, acc11, false, false);
        }
        __syncthreads();
    }

    float* omb = om + (size_t)b * (NOFF * HW) + pixBase + waveId * 32;
    const int mAdd = (lane < 16) ? 0 : 8;
#pragma unroll
    for (int r = 0; r < 8; ++r) {
        int j = r + mAdd;
        if (j < NOFF) {
            omb[j * HW + nlan]      = acc00[r] + b_off[j];
            omb[j * HW + 16 + nlan] = acc01[r] + b_off[j];
        }
        int j2 = 16 + r + mAdd;
        if (j2 < NOFF) {
            omb[j2 * HW + nlan]      = acc10[r] + b_off[j2];
            omb[j2 * HW + 16 + nlan] = acc11[r] + b_off[j2];
        }
    }
}

// ---------------------------------------------------------------------------
// Kernel 2: fused bilinear-sample + GEMM (fp32 WMMA 16x16x4) + BN + ReLU.
// A-chunk staged by the Tensor Data Mover (wave 0 issues, TENSORcnt-tracked),
// overlapped with the bilinear gather phase done by all 8 waves.
// ---------------------------------------------------------------------------
__global__ __launch_bounds__(256) void dcn_main_wmma(
    const float* __restrict__ x, const float* __restrict__ om,
    const float* __restrict__ w_conv, const float* __restrict__ b_conv,
    const float* __restrict__ gamma, const float* __restrict__ beta,
    const float* __restrict__ run_mean, const float* __restrict__ run_var,
    float* __restrict__ out)
{
    __shared__ __align__(16) float A_lds[O_OUT * KSTR];  // 36864 B (TDM dest)
    __shared__ float V_lds[NPIX * KSTR];        //  4608 B : sampled vals [pix][K]
    __shared__ float sw[4][9 * NPIX];           //  4608 B
    __shared__ int   soff[4][9 * NPIX];         //  4608 B
    __shared__ float s_scale[O_OUT];            //  1024 B
    __shared__ float s_shift[O_OUT];            //  1024 B

    const int blocksPerImg = HW / NPIX;         // 128
    const int b   = blockIdx.x / blocksPerImg;
    const int p0  = (blockIdx.x % blocksPerImg) * NPIX;
    const int tid = threadIdx.x;
    const int lane   = tid & 31;
    const int waveId = tid >> 5;
    const unsigned A_base = (unsigned)(uintptr_t)(void*)&A_lds[0]; // LDS addr[31:0]

    // BN folding: out = gemm*scale + shift
    for (int o = tid; o < O_OUT; o += 256) {
        float inv = gamma[o] * rsqrtf(run_var[o] + 1e-5f);
        s_scale[o] = inv;
        s_shift[o] = beta[o] + (b_conv[o] - run_mean[o]) * inv;
    }

    // Sampling params: mask/validity-folded bilinear weights + clamped offsets
    const float* omb = om + (size_t)b * (NOFF * HW);
    for (int i = tid; i < 9 * NPIX; i += 256) {
        int k  = i / NPIX;
        int p  = i - k * NPIX;
        int pp = p0 + p;
        int py = pp >> 6, px = pp & 63;
        float dy   = omb[(2 * k) * HW + pp];
        float dx   = omb[(2 * k + 1) * HW + pp];
        float mraw = omb[(18 + k) * HW + pp];
        float msk  = 1.0f / (1.0f + __expf(-mraw));
        float sy = (float)py - 1.0f + (float)(k / 3) + dy;
        float sx = (float)px - 1.0f + (float)(k % 3) + dx;
        float fy = floorf(sy), fx = floorf(sx);
        int   y0 = (int)fy,    x0 = (int)fx;
        float ly = sy - fy,    lx = sx - fx;
        float hy = 1.0f - ly,  hx = 1.0f - lx;
        int y1 = y0 + 1, x1 = x0 + 1;
        float vy0 = (y0 >= 0 && y0 < H_DIM) ? 1.0f : 0.0f;
        float vy1 = (y1 >= 0 && y1 < H_DIM) ? 1.0f : 0.0f;
        float vx0 = (x0 >= 0 && x0 < W_DIM) ? 1.0f : 0.0f;
        float vx1 = (x1 >= 0 && x1 < W_DIM) ? 1.0f : 0.0f;
        int cy0 = min(max(y0, 0), H_DIM - 1), cy1 = min(max(y1, 0), H_DIM - 1);
        int cx0 = min(max(x0, 0), W_DIM - 1), cx1 = min(max(x1, 0), W_DIM - 1);
        sw[0][i] = hy * hx * msk * vy0 * vx0;
        sw[1][i] = hy * lx * msk * vy0 * vx1;
        sw[2][i] = ly * hx * msk * vy1 * vx0;
        sw[3][i] = ly * lx * msk * vy1 * vx1;
        soff[0][i] = cy0 * W_DIM + cx0;
        soff[1][i] = cy0 * W_DIM + cx1;
        soff[2][i] = cy1 * W_DIM + cx0;
        soff[3][i] = cy1 * W_DIM + cx1;
    }
    __syncthreads();

    v8f acc00 = {}, acc01 = {}, acc10 = {}, acc11 = {};
    const int m0 = waveId * 32;
    const float* xb = x + (size_t)b * (C_IN * HW);

    const int ksel = (lane < 16) ? 0 : 2;
    const int nlan = lane & 15;

    for (int cBase = 0; cBase < C_IN; cBase += 4) {
        // Kick off the A-tile DMA (wave 0), then everyone gathers V while it runs.
        if (waveId == 0)
            tdm_load_A_tile(w_conv + cBase * 9, A_base);

        for (int j = tid; j < KC * NPIX; j += 256) {
            int kk = j >> 5;
            int p  = j & 31;
            int c  = cBase + kk / 9;
            int k  = kk - (kk / 9) * 9;
            int si = k * NPIX + p;
            const float* xc = xb + c * HW;
            float v = sw[0][si] * xc[soff[0][si]]
                    + sw[1][si] * xc[soff[1][si]]
                    + sw[2][si] * xc[soff[2][si]]
                    + sw[3][si] * xc[soff[3][si]];
            V_lds[p * KSTR + kk] = v;
        }
        if (waveId == 0)
            __builtin_amdgcn_s_wait_tensorcnt(0);
        __syncthreads();

#pragma unroll
        for (int kk = 0; kk < KC; kk += 4) {
            v2f a0 = *(const v2f*)&A_lds[(m0 + nlan)      * KSTR + kk + ksel];
            v2f a1 = *(const v2f*)&A_lds[(m0 + 16 + nlan) * KSTR + kk + ksel];
            v2f b0 = *(const v2f*)&V_lds[(nlan)      * KSTR + kk + ksel];
            v2f b1 = *(const v2f*)&V_lds[(16 + nlan) * KSTR + kk + ksel];
            acc00 = __builtin_amdgcn_wmma_f32_16x16x4_f32(false, a0, false, b0, (short)0, acc00, false, false);
            acc01 = __builtin_amdgcn_wmma_f32_16x16x4_f32(false, a0, false, b1, (short)0, acc01, false, false);
            acc10 = __builtin_amdgcn_wmma_f32_16x16x4_f32(false, a1, false, b0, (short)0, acc10, false, false);
            acc11 = __builtin_amdgcn_wmma_f32_16x16x4_f32(false, a1, false, b1, (short)0, acc11, false, false);
        }
        __syncthreads();
    }

    // Epilogue: BN + ReLU
    float* outb = out + (size_t)b * (O_OUT * HW) + p0;
    const int mAdd = (lane < 16) ? 0 : 8;
#pragma unroll
    for (int r = 0; r < 8; ++r) {
        int o = m0 + r + mAdd;
        float v;
        v = acc00[r] * s_scale[o] + s_shift[o];
        outb[o * HW + nlan] = fmaxf(v, 0.0f);
        v = acc01[r] * s_scale[o] + s_shift[o];
        outb[o * HW + 16 + nlan] = fmaxf(v, 0.0f);
        o = m0 + 16 + r + mAdd;
        v = acc10[r] * s_scale[o] + s_shift[o];
        outb[o * HW + nlan] = fmaxf(v, 0.0f);
        v = acc11[r] * s_scale[o] + s_shift[o];
        outb[o * HW + 16 + nlan] = fmaxf(v, 0.0f);
    }
}

// ---------------------------------------------------------------------------
extern "C" void kernel_launch(void* const* d_in, const int* in_sizes, int n_in,
                              void* d_out, int out_size, void* d_ws, size_t ws_size,
                              hipStream_t stream) {
    const float* x        = (const float*)d_in[0];
    const float* w_off    = (const float*)d_in[1];
    const float* b_off    = (const float*)d_in[2];
    const float* w_conv   = (const float*)d_in[3];
    const float* b_conv   = (const float*)d_in[4];
    const float* gamma    = (const float*)d_in[5];
    const float* beta     = (const float*)d_in[6];
    const float* run_mean = (const float*)d_in[7];
    const float* run_var  = (const float*)d_in[8];
    float* out = (float*)d_out;
    float* om  = (float*)d_ws;   // B*27*HW floats = 3.54 MB

    dcn_offset_conv_wmma<<<B_IMG * 16, 256, 0, stream>>>(x, w_off, b_off, om);
    dcn_main_wmma<<<B_IMG * (HW / NPIX), 256, 0, stream>>>(
        x, om, w_conv, b_conv, gamma, beta, run_mean, run_var, out);
}